// AutoCorrelationLoss_9131100471358
// MI455X (gfx1250) — compile-verified
//
#include <hip/hip_runtime.h>

// AutoCorrelationLoss on MI455X (gfx1250, wave32).
//
// ac[k] = (S[k] - m0*T[k])/com with S[k]=sum_t x[t]x[t+k], T[k]=sum_t x[t+k],
// m0=T[0]/com; ac[k]/ac[0] = (S[k]-m0*T[k])/(S[0]-m0*T[0]).
//
// S[k] computed with V_WMMA_F32_16X16X4_F32 (full f32 precision — problem is
// bandwidth/latency-trivial at 2 MB total, so no reason to drop to f16):
//   A[m][kk] = x[t+kk]            (broadcast over M, per ISA 16x4 f32 layout)
//   B[kk][n] = x[t+kk+k0+n]       (16 lags in N)
//   D[m][n] += sum_kk A*B         (chain of accumulating WMMAs over t)
//
// Row staging uses CDNA5 GLOBAL_LOAD_ASYNC_TO_LDS_B128 (ASYNCcnt path, no
// VGPR round-trip); LDS row is padded by 16 zeros so the B fetch needs no
// per-step clamping (lags 129..143 are computed into padding and discarded).

#define LROW   8192
#define LPAD   (LROW + 16)
#define NCO    128
#define COM    (LROW - NCO)        // 8064
#define NT     144                 // 9 lag tiles of 16
#define TILES  9
#define WAVES  8
#define TSLICE (COM / WAVES)       // 1008

typedef __attribute__((ext_vector_type(2))) float v2f;
typedef __attribute__((ext_vector_type(8))) float v8f;

__global__ __launch_bounds__(256)
void corr_sums_kernel(const float* __restrict__ fake,
                      const float* __restrict__ realp,
                      float* __restrict__ st)   // [64][2*NT]: S[0..143], T[0..143]
{
    __shared__ float xrow[LPAD];               // 32 KB row stage + 64 B pad
    __shared__ float sred[WAVES][16];
    __shared__ float tred[WAVES][16];

    const int row  = blockIdx.x;               // 0..63 (0..31 fake, 32..63 real)
    const int k0   = blockIdx.y * 16;          // lag tile base
    const int tid  = threadIdx.x;

    const float* src = (row < 32) ? (fake  + (size_t)row * LROW)
                                  : (realp + (size_t)(row - 32) * LROW);

    // Async global->LDS staging (16B per op, tracked by ASYNCcnt).
    for (int i = tid; i < LROW / 4; i += 256) {
        unsigned           lds_off = (unsigned)(size_t)(const void*)&xrow[i * 4];
        unsigned long long gaddr   = (unsigned long long)(size_t)(src + i * 4);
        asm volatile("global_load_async_to_lds_b128 %0, %1, off"
                     :: "v"(lds_off), "v"(gaddr) : "memory");
    }
    if (tid < 16) xrow[LROW + tid] = 0.0f;     // zero pad (normal DS store)
    asm volatile("s_wait_asynccnt 0" ::: "memory");
    __syncthreads();

    const int wave   = tid >> 5;
    const int lane   = tid & 31;
    const int lane16 = lane & 15;
    const int off    = (lane >= 16) ? 2 : 0;   // hi half-wave supplies K=2,3

    v8f   acc   = {};
    float tpart = 0.0f;

    const int t0 = wave * TSLICE;
    const int t1 = t0 + TSLICE;
    for (int t = t0; t < t1; t += 4) {
        // A: x0 chunk broadcast across M (lane16-invariant -> LDS broadcast,
        //    t+off is even -> 8B aligned -> single ds_load_b64)
        v2f a;
        a.x = xrow[t + off];
        a.y = xrow[t + off + 1];
        // B: lagged chunk, 16 consecutive lags across N (conflict-free);
        //    max index 8206 < LPAD, no clamp needed.
        const int bi = t + off + k0 + lane16;
        v2f b;
        b.x = xrow[bi];
        b.y = xrow[bi + 1];
        // D = A x B + D  (f32, 16x16x4)
        acc = __builtin_amdgcn_wmma_f32_16x16x4_f32(false, a, false, b,
                                                    (short)0, acc, false, false);
        tpart += b.x + b.y;                    // rides along for T[k]
    }
    // T partial: lo lanes covered kk=0,1; hi lanes kk=2,3 — combine halves.
    tpart += __shfl_down(tpart, 16, 32);

    if (lane < 16) {
        // D row M=0 (lanes 0..15, VGPR0) = S partial for lag k0+lane16.
        sred[wave][lane16] = acc[0];
        tred[wave][lane16] = tpart;
    }
    __syncthreads();

    if (tid < 16) {
        float s = 0.0f, tt = 0.0f;
        #pragma unroll
        for (int w = 0; w < WAVES; ++w) { s += sred[w][tid]; tt += tred[w][tid]; }
        float* out = st + (size_t)row * (2 * NT);
        out[k0 + tid]      = s;
        out[NT + k0 + tid] = tt;
    }
}

__global__ __launch_bounds__(256)
void finalize_kernel(const float* __restrict__ st, float* __restrict__ out)
{
    __shared__ float red[256];
    const int tid = threadIdx.x;
    const float invcom = 1.0f / (float)COM;

    float sum = 0.0f;
    for (int item = tid; item < 32 * (NCO + 1); item += 256) {
        const int b = item / (NCO + 1);
        const int k = item - b * (NCO + 1);
        const float* f = st + (size_t)b        * (2 * NT);
        const float* r = st + (size_t)(b + 32) * (2 * NT);
        const float m0f = f[NT] * invcom;
        const float fa  = (f[k] - m0f * f[NT + k]) / (f[0] - m0f * f[NT]);
        const float m0r = r[NT] * invcom;
        const float ra  = (r[k] - m0r * r[NT + k]) / (r[0] - m0r * r[NT]);
        sum += fabsf(fa - ra);
    }
    red[tid] = sum;
    __syncthreads();
    for (int s = 128; s > 0; s >>= 1) {
        if (tid < s) red[tid] += red[tid + s];
        __syncthreads();
    }
    if (tid == 0) out[0] = red[0] / (32.0f * (float)(NCO + 1));
}

extern "C" void kernel_launch(void* const* d_in, const int* in_sizes, int n_in,
                              void* d_out, int out_size, void* d_ws, size_t ws_size,
                              hipStream_t stream) {
    const float* fake  = (const float*)d_in[0];
    const float* realp = (const float*)d_in[1];
    float* st = (float*)d_ws;                  // 64 * 288 f32 = 73,728 B

    dim3 grid(64, TILES);
    corr_sums_kernel<<<grid, 256, 0, stream>>>(fake, realp, st);
    finalize_kernel<<<1, 256, 0, stream>>>(st, (float*)d_out);
}